// NetVLAD_wrapper_46420006535578
// MI455X (gfx1250) — compile-verified
//
#include <hip/hip_runtime.h>
#include <hip/hip_bf16.h>
#include <math.h>
#include <stdint.h>

typedef float v2f __attribute__((ext_vector_type(2)));
typedef float v8f __attribute__((ext_vector_type(8)));

#define B   16
#define C   256
#define N   16384
#define K   64
#define CO  512
#define S   16                 // splits of N per batch
#define NCHUNK (N / S)         // 1024 columns per workgroup
#define NT  32                 // columns per tile iteration
#define TITER (NCHUNK / NT)    // 32 iterations
#define XS  36                 // LDS stride for x tile (144B: 16B-aligned rows)
#define AS  33                 // LDS stride for assignment tile (pad)
#define WSR 257                // LDS stride for W (pad)
#define EPSN 1e-12f
#define BN_EPS 1e-5f

__device__ __forceinline__ v8f wmma_f32_k4(v2f a, v2f b, v8f c) {
    return __builtin_amdgcn_wmma_f32_16x16x4_f32(
        false, a, false, b, (short)0, c, false, false);
}

// Async global->LDS copy of one 256xNT x-tile (b128: 8 lanes cover a 128B row).
__device__ __forceinline__ void issue_tile_async(const float* __restrict__ xb,
                                                 int n0, float* ldsbuf, int tid) {
    const int j4 = (tid & 7) * 4;      // column group within tile (16B)
    const int cr = tid >> 3;           // 0..31 base row
    for (int r = 0; r < 8; ++r) {
        const int c = cr + 32 * r;
        const uint64_t gaddr = (uint64_t)(uintptr_t)(xb + (size_t)c * N + n0 + j4);
        const uint32_t laddr = (uint32_t)(uintptr_t)(ldsbuf + c * XS + j4);
        asm volatile("global_load_async_to_lds_b128 %0, %1, off"
                     :: "v"(laddr), "v"(gaddr) : "memory");
    }
}

__device__ __forceinline__ void wait_async0() {
    asm volatile("s_wait_asynccnt 0x0" ::: "memory");
}

// ---------------- Kernel 1: fused norm + assign GEMM + softmax + agg GEMM ----
__global__ __launch_bounds__(256)
void k1_fused(const float* __restrict__ x,
              const float* __restrict__ assign_w,
              const float* __restrict__ assign_b,
              float* __restrict__ aggP,     // (B*S, K, C) partials
              float* __restrict__ asumP) {  // (B*S, K) partials
    __shared__ float sW[K * WSR];          // 64x257
    __shared__ float sX[2][C * XS];        // double-buffered 256x36
    __shared__ float sA[K * AS];           // 64x33
    __shared__ float sBias[K];
    __shared__ float sRed[8 * NT];
    __shared__ float sInv[NT];
    __shared__ float sCol[NT];

    const int blk = blockIdx.x;
    const int b = blk / S;
    const int s = blk % S;
    const int n_base = s * NCHUNK;
    const int tid = threadIdx.x;
    const int w = tid >> 5;            // wave id 0..7 (wave32)
    const int lane = tid & 31;
    const int l16 = lane & 15;
    const int half = lane >> 4;
    const int khalf = half << 1;
    const int mt = w & 3;              // K-row tile for GEMMs
    const int nthalf = w >> 2;         // 0..1

    // stage assign_w and bias into LDS
    for (int i = tid; i < K * C; i += 256) {
        sW[(i / C) * WSR + (i % C)] = assign_w[i];
    }
    if (tid < K) sBias[tid] = assign_b[tid];

    v8f acc2[8];
    for (int i = 0; i < 8; ++i) acc2[i] = (v8f){};
    float asum_reg = 0.0f;

    const float* xb = x + (size_t)b * C * N;
    const int j = tid & 31;
    const int crow = tid >> 5;

    // kick off async copy of tile 0
    issue_tile_async(xb, n_base, &sX[0][0], tid);
    __syncthreads();

    for (int t = 0; t < TITER; ++t) {
        float* xt = &sX[t & 1][0];
        // wave-local async completion, then workgroup visibility
        wait_async0();
        __syncthreads();
        // overlap: start next tile's global->LDS DMA while computing this one
        if (t + 1 < TITER)
            issue_tile_async(xb, n_base + (t + 1) * NT, &sX[(t + 1) & 1][0], tid);

        // ---- column L2 norms over C ----
        {
            float ss = 0.0f;
            const int cb = crow * 32;
            for (int i = 0; i < 32; ++i) {
                float v = xt[(cb + i) * XS + j];
                ss += v * v;
            }
            sRed[crow * NT + j] = ss;
        }
        __syncthreads();
        if (tid < NT) {
            float tot = 0.0f;
            for (int g = 0; g < 8; ++g) tot += sRed[g * NT + tid];
            sInv[tid] = 1.0f / fmaxf(sqrtf(tot), EPSN);
        }
        __syncthreads();
        {
            const float inv = sInv[j];
            for (int r = 0; r < C; r += 8) xt[(crow + r) * XS + j] *= inv;
        }
        __syncthreads();
        // ---- GEMM1: logits(64xNT) = W(64xC) * xn(CxNT); wave owns 16x16 tile
        {
            v8f acc1 = (v8f){};
            for (int c0 = 0; c0 < C; c0 += 4) {
                v2f a, bv;
                a.x = sW[(mt * 16 + l16) * WSR + c0 + khalf];
                a.y = sW[(mt * 16 + l16) * WSR + c0 + khalf + 1];
                bv.x = xt[(c0 + khalf) * XS + nthalf * 16 + l16];
                bv.y = xt[(c0 + khalf + 1) * XS + nthalf * 16 + l16];
                acc1 = wmma_f32_k4(a, bv, acc1);
            }
            for (int i = 0; i < 8; ++i) {
                const int kk = mt * 16 + i + 8 * half;
                sA[kk * AS + nthalf * 16 + l16] = acc1[i] + sBias[kk];
            }
        }
        __syncthreads();
        // ---- softmax over K per column ----
        {
            float m = -3.4e38f;
            for (int i = 0; i < 8; ++i) m = fmaxf(m, sA[(crow + 8 * i) * AS + j]);
            sRed[crow * NT + j] = m;
        }
        __syncthreads();
        if (tid < NT) {
            float mm = -3.4e38f;
            for (int q = 0; q < 8; ++q) mm = fmaxf(mm, sRed[q * NT + tid]);
            sCol[tid] = mm;
        }
        __syncthreads();
        {
            const float cm = sCol[j];
            float ssum = 0.0f;
            for (int i = 0; i < 8; ++i) {
                const int kk = crow + 8 * i;
                float e = __expf(sA[kk * AS + j] - cm);
                sA[kk * AS + j] = e;
                ssum += e;
            }
            sRed[crow * NT + j] = ssum;
        }
        __syncthreads();
        if (tid < NT) {
            float tot = 0.0f;
            for (int q = 0; q < 8; ++q) tot += sRed[q * NT + tid];
            sCol[tid] = 1.0f / tot;
        }
        __syncthreads();
        {
            const float ci = sCol[j];
            for (int i = 0; i < 8; ++i) sA[(crow + 8 * i) * AS + j] *= ci;
        }
        __syncthreads();
        // ---- a_sum accumulation (register, per k) ----
        if (tid < K) {
            float s2 = 0.0f;
            for (int jj = 0; jj < NT; ++jj) s2 += sA[tid * AS + jj];
            asum_reg += s2;
        }
        // ---- GEMM2: agg(64x256) += a(64xNT) * xn^T(NTx256); wave owns 8 tiles
        for (int i = 0; i < 8; ++i) {
            const int nt2 = nthalf * 8 + i;       // c-tile 0..15
            v8f acc = acc2[i];
            for (int k0 = 0; k0 < NT; k0 += 4) {
                v2f a, bv;
                a.x = sA[(mt * 16 + l16) * AS + k0 + khalf];
                a.y = sA[(mt * 16 + l16) * AS + k0 + khalf + 1];
                bv.x = xt[(nt2 * 16 + l16) * XS + k0 + khalf];
                bv.y = xt[(nt2 * 16 + l16) * XS + k0 + khalf + 1];
                acc = wmma_f32_k4(a, bv, acc);
            }
            acc2[i] = acc;
        }
        __syncthreads();   // buffer (t&1) free for reuse by iteration t+2's DMA
    }

    // ---- write partials ----
    float* aggp = aggP + (size_t)(b * S + s) * K * C;
    for (int i = 0; i < 8; ++i) {
        const int nt2 = nthalf * 8 + i;
        for (int r = 0; r < 8; ++r) {
            const int kk = mt * 16 + r + 8 * half;
            const int cc = nt2 * 16 + l16;
            aggp[kk * C + cc] = acc2[i][r];
        }
    }
    if (tid < K) asumP[(size_t)(b * S + s) * K + tid] = asum_reg;
}

// ---------------- Kernel 2: reduce splits, subtract centroids, intra-norm ----
__global__ __launch_bounds__(256)
void k2_vlad(const float* __restrict__ aggP,
             const float* __restrict__ asumP,
             const float* __restrict__ centroids,
             float* __restrict__ vladB,    // (B, K*C) intra-normalized
             float* __restrict__ rowSq) {  // (B, K) sum of squares after norm
    __shared__ float red[256];
    const int b = blockIdx.x / K;
    const int k = blockIdx.x % K;
    const int c = threadIdx.x;

    float sum = 0.0f;
    for (int s = 0; s < S; ++s)
        sum += aggP[((size_t)(b * S + s) * K + k) * C + c];
    float as = 0.0f;
    for (int s = 0; s < S; ++s)
        as += asumP[(size_t)(b * S + s) * K + k];

    const float v = sum - as * centroids[k * C + c];
    red[c] = v * v;
    __syncthreads();
    for (int off = 128; off > 0; off >>= 1) {
        if (c < off) red[c] += red[c + off];
        __syncthreads();
    }
    const float ssq = red[0];
    const float inv = 1.0f / fmaxf(sqrtf(ssq), EPSN);
    vladB[(size_t)b * (K * C) + k * C + c] = v * inv;
    if (c == 0) rowSq[b * K + k] = ssq * inv * inv;
}

// ---------------- Kernel 3: per-batch global L2 scale ----------------------
__global__ __launch_bounds__(64)
void k3_gnorm(const float* __restrict__ rowSq, float* __restrict__ gscale) {
    __shared__ float red[64];
    const int b = blockIdx.x;
    const int t = threadIdx.x;
    red[t] = rowSq[b * K + t];
    __syncthreads();
    for (int off = 32; off > 0; off >>= 1) {
        if (t < off) red[t] += red[t + off];
        __syncthreads();
    }
    if (t == 0) gscale[b] = 1.0f / fmaxf(sqrtf(red[0]), EPSN);
}

// ---------------- Kernel 4: reduction GEMM y = g * vlad @ red_w^T + red_b ---
__global__ __launch_bounds__(32)
void k4_reduct(const float* __restrict__ vladB,
               const float* __restrict__ red_w,
               const float* __restrict__ red_b,
               const float* __restrict__ gscale,
               float* __restrict__ y) {   // (B, CO)
    const int cot = blockIdx.x;            // CO/16 tiles
    const int lane = threadIdx.x;
    const int l16 = lane & 15;
    const int half = lane >> 4;
    const int khalf = half << 1;

    v8f acc = (v8f){};
    for (int k0 = 0; k0 < K * C; k0 += 4) {
        v2f a, bv;
        a.x = vladB[(size_t)l16 * (K * C) + k0 + khalf];
        a.y = vladB[(size_t)l16 * (K * C) + k0 + khalf + 1];
        bv.x = red_w[(size_t)(cot * 16 + l16) * (K * C) + k0 + khalf];
        bv.y = red_w[(size_t)(cot * 16 + l16) * (K * C) + k0 + khalf + 1];
        acc = wmma_f32_k4(a, bv, acc);
    }
    for (int i = 0; i < 8; ++i) {
        const int m = i + 8 * half;        // batch index
        const int co = cot * 16 + l16;
        y[m * CO + co] = acc[i] * gscale[m] + red_b[co];
    }
}

// ---------------- Kernel 5: BatchNorm (batch stats) + ReLU ------------------
__global__ __launch_bounds__(256)
void k5_bnrelu(const float* __restrict__ y,
               const float* __restrict__ bn_w,
               const float* __restrict__ bn_b,
               float* __restrict__ out) {  // (B, CO, 1) flat
    const int co = blockIdx.x * 256 + threadIdx.x;
    float m = 0.0f;
    for (int b = 0; b < B; ++b) m += y[b * CO + co];
    m *= (1.0f / B);
    float var = 0.0f;
    for (int b = 0; b < B; ++b) {
        const float d = y[b * CO + co] - m;
        var += d * d;
    }
    var *= (1.0f / B);
    const float is = rsqrtf(var + BN_EPS);
    const float gw = bn_w[co], gb = bn_b[co];
    for (int b = 0; b < B; ++b) {
        const float v = gw * (y[b * CO + co] - m) * is + gb;
        out[b * CO + co] = fmaxf(v, 0.0f);
    }
}

extern "C" void kernel_launch(void* const* d_in, const int* in_sizes, int n_in,
                              void* d_out, int out_size, void* d_ws, size_t ws_size,
                              hipStream_t stream) {
    const float* x         = (const float*)d_in[0];
    const float* assign_w  = (const float*)d_in[1];
    const float* assign_b  = (const float*)d_in[2];
    const float* centroids = (const float*)d_in[3];
    const float* red_w     = (const float*)d_in[4];
    const float* red_b     = (const float*)d_in[5];
    const float* bn_w      = (const float*)d_in[6];
    const float* bn_b      = (const float*)d_in[7];
    float* out = (float*)d_out;

    float* wsf    = (float*)d_ws;
    float* aggP   = wsf;                                   // B*S*K*C
    float* asumP  = aggP  + (size_t)B * S * K * C;         // B*S*K
    float* vladB  = asumP + (size_t)B * S * K;             // B*K*C
    float* rowSq  = vladB + (size_t)B * K * C;             // B*K
    float* gscale = rowSq + (size_t)B * K;                 // B
    float* yBuf   = gscale + B;                            // B*CO

    k1_fused <<<dim3(B * S), dim3(256), 0, stream>>>(x, assign_w, assign_b, aggP, asumP);
    k2_vlad  <<<dim3(B * K), dim3(256), 0, stream>>>(aggP, asumP, centroids, vladB, rowSq);
    k3_gnorm <<<dim3(B),     dim3(64),  0, stream>>>(rowSq, gscale);
    k4_reduct<<<dim3(CO/16), dim3(32),  0, stream>>>(vladB, red_w, red_b, gscale, yBuf);
    k5_bnrelu<<<dim3(CO/256),dim3(256), 0, stream>>>(yBuf, bn_w, bn_b, out);
}